// GatedCRFLoss_56779467653466
// MI455X (gfx1250) — compile-verified
//
#include <hip/hip_runtime.h>

// Problem constants (match reference)
#define SPAN 11
#define KW   23           // window width, K*K = 529 offsets
#define NB   2
#define NC   3
#define NH   160
#define NW   160
#define NPIX (NH * NW)
#define TILE 16
#define PH   (TILE + 2 * SPAN)   // 38 padded tile width

#define LOG2E   1.4426950408889634f
#define NEGHALF_LOG2E 0.7213475204444817f   // 0.5*log2(e); exp(-0.5x)=2^(-this*x)
#define LN2     0.6931471805599453f

typedef float v2f __attribute__((ext_vector_type(2)));
typedef float v8f __attribute__((ext_vector_type(8)));

// ---------------------------------------------------------------------------
// Kernel 0: zero the two global accumulators (ce_sum, gcrf_sum)
// ---------------------------------------------------------------------------
__global__ void zero_accum_kernel(float* accum) {
    accum[0] = 0.0f;
    accum[1] = 0.0f;
}

// ---------------------------------------------------------------------------
// Kernel 1: LDS-tiled CRF stencil.
//   grid = (W/16, H/16, B), block = 256 (16x16 pixels, 8 wave32)
//   Writes Egy (B,3,H,W) and y_hat (B,3,H,W) to workspace,
//   atomically accumulates the CE sum into accum[0].
// ---------------------------------------------------------------------------
__global__ __launch_bounds__(256) void gcrf_main_kernel(
    const float* __restrict__ logit,
    const int*   __restrict__ target,
    const float* __restrict__ image,
    const float* __restrict__ src_map,
    const float* __restrict__ dst_map,
    float* __restrict__ egy,
    float* __restrict__ yhat,
    float* __restrict__ accum)
{
    __shared__ float s_img0[PH * PH];
    __shared__ float s_img1[PH * PH];
    __shared__ float s_img2[PH * PH];
    __shared__ float s_yh0[PH * PH];
    __shared__ float s_yh1[PH * PH];
    __shared__ float s_yh2[PH * PH];
    __shared__ float s_src[PH * PH];
    __shared__ float s_f[TILE * KW];   // exp(-0.5*(i - (i+dx)/6)^2) per tile row
    __shared__ float s_g[TILE * KW];   // same for columns
    __shared__ float s_ce;

    const int tid = threadIdx.x;
    const int b   = blockIdx.z;
    const int ty0 = blockIdx.y * TILE;
    const int tx0 = blockIdx.x * TILE;

    if (tid == 0) s_ce = 0.0f;

    const float* imb = image  + (size_t)b * NC * NPIX;
    const float* lgb = logit  + (size_t)b * NC * NPIX;
    const float* smb = src_map + (size_t)b * NPIX;

    // ---- stage padded halo: image (3ch), softmax(logit) (3ch), source map
    for (int p = tid; p < PH * PH; p += 256) {
        const int pr = p / PH;
        const int pc = p - pr * PH;
        const int gi = ty0 + pr - SPAN;
        const int gj = tx0 + pc - SPAN;
        float i0 = 0.f, i1 = 0.f, i2 = 0.f, sm = 0.f;
        float y0 = 0.f, y1 = 0.f, y2 = 0.f;
        if (gi >= 0 && gi < NH && gj >= 0 && gj < NW) {
            const int off = gi * NW + gj;
            i0 = imb[off];
            i1 = imb[NPIX + off];
            i2 = imb[2 * NPIX + off];
            sm = smb[off];
            const float l0 = lgb[off];
            const float l1 = lgb[NPIX + off];
            const float l2 = lgb[2 * NPIX + off];
            const float mx = fmaxf(l0, fmaxf(l1, l2));
            const float e0 = exp2f((l0 - mx) * LOG2E);
            const float e1 = exp2f((l1 - mx) * LOG2E);
            const float e2 = exp2f((l2 - mx) * LOG2E);
            const float inv = 1.0f / (e0 + e1 + e2);
            y0 = e0 * inv; y1 = e1 * inv; y2 = e2 * inv;
        }
        s_img0[p] = i0; s_img1[p] = i1; s_img2[p] = i2;
        s_yh0[p]  = y0; s_yh1[p]  = y1; s_yh2[p]  = y2;
        s_src[p]  = sm;
    }

    // ---- separable spatial kernel tables (faithful to reference "bug":
    //      u = i - (i+dx)/SIG_XY, not a pure delta)
    for (int t = tid; t < TILE * KW; t += 256) {
        const int r  = t / KW;
        const int d  = t - r * KW;
        const int dx = d - SPAN;
        const float fi = (float)(ty0 + r);
        const float u  = fi - (float)(ty0 + r + dx) * (1.0f / 6.0f);
        s_f[t] = exp2f(-NEGHALF_LOG2E * u * u);
        const float fj = (float)(tx0 + r);
        const float v  = fj - (float)(tx0 + r + dx) * (1.0f / 6.0f);
        s_g[t] = exp2f(-NEGHALF_LOG2E * v * v);
    }
    __syncthreads();

    // ---- per-pixel window accumulation
    const int ty = tid >> 4;
    const int tx = tid & 15;
    const int ci = (ty + SPAN) * PH + (tx + SPAN);
    const float c0 = s_img0[ci], c1 = s_img1[ci], c2 = s_img2[ci];

    float e0 = 0.f, e1 = 0.f, e2 = 0.f;
    for (int di = 0; di < KW; ++di) {
        const int   rb = (ty + di) * PH + tx;
        const float fr = s_f[ty * KW + di];
        for (int dj = 0; dj < KW; ++dj) {
            const int   q  = rb + dj;
            const float ms = s_src[q];                 // 0 for OOB and ungated
            const float d0 = c0 - 10.0f * s_img0[q];   // center - shifted/SIG_RGB
            const float d1 = c1 - 10.0f * s_img1[q];
            const float d2 = c2 - 10.0f * s_img2[q];
            const float r2 = d0 * d0 + d1 * d1 + d2 * d2;
            const float krgb = exp2f(-NEGHALF_LOG2E * r2);
            const float kg = (krgb + fr * s_g[tx * KW + dj]) * ms;
            e0 = fmaf(s_yh0[q], kg, e0);
            e1 = fmaf(s_yh1[q], kg, e1);
            e2 = fmaf(s_yh2[q], kg, e2);
        }
    }

    const int gi  = ty0 + ty;
    const int gj  = tx0 + tx;
    const int off = gi * NW + gj;
    const float dmap = dst_map[(size_t)b * NPIX + off];
    const float md = 1.0f - dmap;                      // m_dst gate

    float* eb = egy  + (size_t)b * NC * NPIX;
    float* yb = yhat + (size_t)b * NC * NPIX;
    eb[off]            = e0 * md;
    eb[NPIX + off]     = e1 * md;
    eb[2 * NPIX + off] = e2 * md;
    yb[off]            = s_yh0[ci];
    yb[NPIX + off]     = s_yh1[ci];
    yb[2 * NPIX + off] = s_yh2[ci];

    // ---- cross-entropy contribution: -log_softmax(logit)[target] * dest_map
    const float l0 = lgb[off];
    const float l1 = lgb[NPIX + off];
    const float l2 = lgb[2 * NPIX + off];
    const float mx = fmaxf(l0, fmaxf(l1, l2));
    const float ssum = exp2f((l0 - mx) * LOG2E) + exp2f((l1 - mx) * LOG2E) +
                       exp2f((l2 - mx) * LOG2E);
    const int   tc = target[(size_t)b * NPIX + off];
    const float lt = (tc == 0) ? l0 : ((tc == 1) ? l1 : l2);
    const float logp = (lt - mx) - log2f(ssum) * LN2;
    atomicAdd(&s_ce, -logp * dmap);
    __syncthreads();
    if (tid == 0) atomicAdd(&accum[0], s_ce);
}

// ---------------------------------------------------------------------------
// Kernel 2: WMMA Gram reduction.
//   temp[b] = Egy(3 x NPIX) * yhat^T(NPIX x 3) via V_WMMA_F32_16X16X4_F32,
//   K = pixel dimension in chunks of 4, full f32 accumulation.
//   Off-diagonal (Moff) masked sum -> accum[1].
//   grid = (B), block = 256 (8 wave32, each strides pixel chunks).
//
// Branch-free operand build: lanes with row >= 3 read through a clamped
// pointer (aliasing row 0) with an unconditional b64 vector load, then
// multiply by a 0/1 mask. No exec-mask churn in the inner loop.
//
// A 16x4 f32 layout: lane = row M (0..15 twice), VGPR v holds K = v + 2*(lane>=16)
// B 4x16 layout mirrors: row K = v + 2*(lane>=16), column N = lane&15
// C/D 16x16: row M = vgpr + 8*(lane>=16), column N = lane&15
// ---------------------------------------------------------------------------
__global__ __launch_bounds__(256) void gram_wmma_kernel(
    const float* __restrict__ egy,
    const float* __restrict__ yhat,
    float* __restrict__ accum)
{
    const int b    = blockIdx.x;
    const int wave = threadIdx.x >> 5;
    const int lane = threadIdx.x & 31;
    const int m    = lane & 15;                 // A row / B column index
    const int kh   = (lane >> 4) << 1;          // K offset: 0 (lo half) or 2 (hi)
    const int   mrow = (m < NC) ? m : 0;        // clamp: inactive rows alias row 0
    const float msk  = (m < NC) ? 1.0f : 0.0f;  // zero out inactive rows

    const float* Ep = egy  + (size_t)b * NC * NPIX + (size_t)mrow * NPIX;
    const float* Yp = yhat + (size_t)b * NC * NPIX + (size_t)mrow * NPIX;

    v8f acc = {0.f, 0.f, 0.f, 0.f, 0.f, 0.f, 0.f, 0.f};

    const int nchunk = NPIX / 4;                // 6400 K-chunks of 4 pixels
    #pragma unroll 4
    for (int ch = wave; ch < nchunk; ch += 8) {
        const int kb = ch * 4 + kh;
        __builtin_prefetch(Ep + kb + 128, 0, 1);   // next K-chunk cachelines
        __builtin_prefetch(Yp + kb + 128, 0, 1);
        v2f a  = *(const v2f*)(Ep + kb);        // unconditional global_load_b64
        v2f bm = *(const v2f*)(Yp + kb);
        a.x *= msk; a.y *= msk;
        bm.x *= msk; bm.y *= msk;
        acc = __builtin_amdgcn_wmma_f32_16x16x4_f32(
            /*neg_a=*/false, a, /*neg_b=*/false, bm,
            /*c_mod=*/(short)0, acc, /*reuse_a=*/false, /*reuse_b=*/false);
    }

    // Moff = 1 - eye(3): sum off-diagonal D[m][n], m,n in 0..2.
    // Hi half lanes hold rows M=8..15 -> all masked out; only lanes 0..2 matter.
    if (lane < NC) {
        float part = 0.0f;
        #pragma unroll
        for (int r = 0; r < 8; ++r) {
            if (r < NC && r != lane) part += acc[r];
        }
        atomicAdd(&accum[1], part);
    }
}

// ---------------------------------------------------------------------------
// Kernel 3: finalize scalar loss
// ---------------------------------------------------------------------------
__global__ void finalize_kernel(const float* __restrict__ accum,
                                float* __restrict__ out)
{
    const float denom = (float)(NB * NPIX);
    out[0] = accum[0] / denom + 0.15f * (accum[1] / denom);
}

// ---------------------------------------------------------------------------
extern "C" void kernel_launch(void* const* d_in, const int* in_sizes, int n_in,
                              void* d_out, int out_size, void* d_ws, size_t ws_size,
                              hipStream_t stream) {
    const float* logit   = (const float*)d_in[0];  // (B,C,H,W) f32
    const int*   target  = (const int*)  d_in[1];  // (B,H,W) i32
    const float* image   = (const float*)d_in[2];  // (B,3,H,W) f32
    const float* src_map = (const float*)d_in[3];  // (B,1,H,W) f32
    const float* dst_map = (const float*)d_in[4];  // (B,H,W) f32
    float* out = (float*)d_out;

    float* ws    = (float*)d_ws;
    float* egy   = ws;                            // B*3*H*W floats
    float* yhat  = ws + (size_t)NB * NC * NPIX;   // B*3*H*W floats
    float* accum = yhat + (size_t)NB * NC * NPIX; // 2 floats

    zero_accum_kernel<<<1, 1, 0, stream>>>(accum);

    dim3 grid(NW / TILE, NH / TILE, NB);          // (10, 10, 2)
    gcrf_main_kernel<<<grid, 256, 0, stream>>>(
        logit, target, image, src_map, dst_map, egy, yhat, accum);

    gram_wmma_kernel<<<NB, 256, 0, stream>>>(egy, yhat, accum);

    finalize_kernel<<<1, 1, 0, stream>>>(accum, out);
}